// LocalConsistencyLoss_45071386804579
// MI455X (gfx1250) — compile-verified
//
#include <hip/hip_runtime.h>
#include <hip/hip_bf16.h>

typedef __attribute__((ext_vector_type(2))) float v2f;
typedef __attribute__((ext_vector_type(8))) float v8f;

#define C_CLASSES 20
#define K_NBR 16
#define LOSS_WEIGHT 1.0f

// ---------------------------------------------------------------------------
// Pass 1: softmax over C=20 per point, store probs (row-major [N][20]) and
// per-row squared L2 norm S[n] = sum(p^2).  Rows are 80B = 5 x float4.
// ---------------------------------------------------------------------------
__global__ void lcl_softmax_kernel(const float* __restrict__ logits,
                                   float* __restrict__ P,
                                   float* __restrict__ S,
                                   int N) {
    int n = blockIdx.x * blockDim.x + threadIdx.x;
    if (n >= N) return;
    const float4* in4 = reinterpret_cast<const float4*>(logits + (size_t)n * C_CLASSES);
    float x[C_CLASSES];
#pragma unroll
    for (int i = 0; i < 5; ++i) {
        float4 v = in4[i];
        x[4*i+0] = v.x; x[4*i+1] = v.y; x[4*i+2] = v.z; x[4*i+3] = v.w;
    }
    float mx = x[0];
#pragma unroll
    for (int c = 1; c < C_CLASSES; ++c) mx = fmaxf(mx, x[c]);
    float s = 0.0f;
#pragma unroll
    for (int c = 0; c < C_CLASSES; ++c) { x[c] = __expf(x[c] - mx); s += x[c]; }
    float inv = 1.0f / s;
    float ss = 0.0f;
    float4* out4 = reinterpret_cast<float4*>(P + (size_t)n * C_CLASSES);
#pragma unroll
    for (int i = 0; i < 5; ++i) {
        float4 v;
        v.x = x[4*i+0] * inv; v.y = x[4*i+1] * inv;
        v.z = x[4*i+2] * inv; v.w = x[4*i+3] * inv;
        ss += v.x*v.x + v.y*v.y + v.z*v.z + v.w*v.w;
        out4[i] = v;
    }
    S[n] = ss;
}

// 16-lane xor-tree sum confined to lanes 0-15 (lanes 16-31 must carry 0).
// ds_swizzle group-of-32 encoding: offset = (xor_mask<<10) | (or<<5) | and_mask
__device__ __forceinline__ float reduce16_xor(float v) {
    v += __int_as_float(__builtin_amdgcn_ds_swizzle(__float_as_int(v), 0x201F)); // xor 8
    v += __int_as_float(__builtin_amdgcn_ds_swizzle(__float_as_int(v), 0x101F)); // xor 4
    v += __int_as_float(__builtin_amdgcn_ds_swizzle(__float_as_int(v), 0x081F)); // xor 2
    v += __int_as_float(__builtin_amdgcn_ds_swizzle(__float_as_int(v), 0x041F)); // xor 1
    return v;
}

// ---------------------------------------------------------------------------
// Pass 2: one wave32 per point.  dot(p, q_k) for the 16 neighbors computed
// with V_WMMA_F32_16X16X4_F32:
//   A (16x4): all 16 rows = p-chunk (lane L/L+16 hold K0,K1 / K2,K3) -> uniform
//   B (4x16): B[c][k] = probs[idx_k][c]; VGPR0/1 hold rows {0,1} for lanes
//             0-15 and rows {2,3} for lanes 16-31 (mirrors C/D half split)
//   D[0][k] (acc VGPR0, lane k) = dot(p, q_k) after 5 accumulating chunks.
// dist = ||p||^2 + ||q||^2 - 2*dot.  Neighbor count via ballot+popcount,
// distance sum via 4-step ds_swizzle xor reduction.
// ---------------------------------------------------------------------------
__global__ void lcl_neighbor_kernel(const int* __restrict__ nbr,
                                    const int* __restrict__ labels,
                                    const float* __restrict__ P,
                                    const float* __restrict__ S,
                                    float* __restrict__ partials,
                                    int N) {
    __shared__ float sl[8];
    __shared__ float sv[8];
    const int lane = threadIdx.x & 31;
    const int wv   = threadIdx.x >> 5;
    const int n    = blockIdx.x * 8 + wv;
    // No divergent branch around WMMA: clamp and zero-weight the tail so EXEC
    // stays all-ones through the matrix ops.
    const float wvalid = (n < N) ? 1.0f : 0.0f;
    const int   nc     = (n < N) ? n : (N - 1);
    // n is wave-uniform: mark it so point-row / norm / label loads scalarize.
    const int   ncu    = __builtin_amdgcn_readfirstlane(nc);

    const int  kL   = lane & 15;
    const int  base = (lane >= 16) ? 2 : 0;

    int   idx_raw = nbr[(size_t)ncu * K_NBR + kL];
    float m       = (idx_raw >= 0) ? 1.0f : 0.0f;
    int   idxc    = idx_raw < 0 ? 0 : (idx_raw > N - 1 ? N - 1 : idx_raw);

    const float* __restrict__ prow = P + (size_t)ncu  * C_CLASSES;
    const float* __restrict__ qrow = P + (size_t)idxc * C_CLASSES;
    __builtin_prefetch(qrow, 0, 3);   // global_prefetch_b8, WGP scope: warm row

    v8f acc = {};
#pragma unroll
    for (int ch = 0; ch < 5; ++ch) {
        v2f a, b;
        a.x = prow[ch * 4 + base];
        a.y = prow[ch * 4 + base + 1];
        b.x = qrow[ch * 4 + base];
        b.y = qrow[ch * 4 + base + 1];
        acc = __builtin_amdgcn_wmma_f32_16x16x4_f32(
            /*neg_a=*/false, a, /*neg_b=*/false, b,
            /*c_mod=*/(short)0, acc, /*reuse_a=*/false, /*reuse_b=*/false);
    }

    float dotpq = acc[0];             // D[0][kL] in lanes 0-15 (dup in 16-31)
    float dist  = S[ncu] + S[idxc] - 2.0f * dotpq;
    float contrib = (lane < 16) ? dist * m : 0.0f;
    contrib = reduce16_xor(contrib);  // lane 0: sum over k of dist*mask

    // valid-neighbor count: lanes 16-31 loaded duplicate indices, so the low
    // 16 ballot bits are exactly the per-k mask.
    unsigned long long bal = __ballot(idx_raw >= 0);
    float cnt = (float)__popcll(bal & 0xFFFFull);

    float vf = ((labels[ncu] != -1) ? 1.0f : 0.0f) * wvalid;
    float pointLoss = (contrib / fmaxf(cnt, 1.0f)) * vf;

    if (lane == 0) { sl[wv] = pointLoss; sv[wv] = vf; }
    __syncthreads();
    if (threadIdx.x == 0) {
        float a = 0.0f, b = 0.0f;
#pragma unroll
        for (int w = 0; w < 8; ++w) { a += sl[w]; b += sv[w]; }
        partials[2 * (size_t)blockIdx.x]     = a;
        partials[2 * (size_t)blockIdx.x + 1] = b;
    }
}

// ---------------------------------------------------------------------------
// Pass 3: deterministic reduction of per-block partials -> scalar loss.
// ---------------------------------------------------------------------------
__global__ void lcl_finalize_kernel(const float* __restrict__ partials,
                                    int nparts,
                                    float* __restrict__ out) {
    __shared__ float sa[256];
    __shared__ float sb[256];
    float a = 0.0f, b = 0.0f;
    for (int i = threadIdx.x; i < nparts; i += 256) {
        a += partials[2 * (size_t)i];
        b += partials[2 * (size_t)i + 1];
    }
    sa[threadIdx.x] = a; sb[threadIdx.x] = b;
    __syncthreads();
    for (int s = 128; s > 0; s >>= 1) {
        if (threadIdx.x < s) {
            sa[threadIdx.x] += sa[threadIdx.x + s];
            sb[threadIdx.x] += sb[threadIdx.x + s];
        }
        __syncthreads();
    }
    if (threadIdx.x == 0) {
        out[0] = (sa[0] / fmaxf(sb[0], 1.0f)) * LOSS_WEIGHT;
    }
}

extern "C" void kernel_launch(void* const* d_in, const int* in_sizes, int n_in,
                              void* d_out, int out_size, void* d_ws, size_t ws_size,
                              hipStream_t stream) {
    const float* logits = (const float*)d_in[0];   // [N, 20] f32
    const int*   nbr    = (const int*)d_in[1];     // [N, 16] int
    const int*   labels = (const int*)d_in[2];     // [N]     int
    const int N = in_sizes[2];                     // labels count == N

    // ws layout: probs [N*20] f32 | norms [N] f32 | partials [2 * nBlocks2] f32
    float* P        = (float*)d_ws;
    float* S        = P + (size_t)N * C_CLASSES;
    float* partials = S + (size_t)N;

    const int b1 = (N + 255) / 256;
    lcl_softmax_kernel<<<b1, 256, 0, stream>>>(logits, P, S, N);

    const int b2 = (N + 7) / 8;   // 8 points (8 waves) per 256-thread block
    lcl_neighbor_kernel<<<b2, 256, 0, stream>>>(nbr, labels, P, S, partials, N);

    lcl_finalize_kernel<<<1, 256, 0, stream>>>(partials, b2, (float*)d_out);
}